// SpectralConvLayer_34084860461078
// MI455X (gfx1250) — compile-verified
//
#include <hip/hip_runtime.h>
#include <math.h>

// Problem constants (from reference): B=32, CENT=2, IN_CH=8, OUT_CH=16, NDIAG=64
#define ND      64
#define CN      16                 // CENT*IN_CH
#define M_GEMM  512                // B*OUT_CH
#define K_GEMM  1024               // CN*ND
#define N_GEMM  4096               // ND*ND

typedef float v2f __attribute__((ext_vector_type(2)));
typedef float v8f __attribute__((ext_vector_type(8)));

// ---------------------------------------------------------------------------
// Kernel 1: one-sided (Hestenes) Jacobi SVD — left singular vectors only.
// One block per (c,n) matrix. Columns of C[c,n] live in LDS; after
// orthogonalization, column norms are the singular values, normalized columns
// are u. Columns are permuted into descending-singular-value order to match
// jnp.linalg.svd (sign ambiguity cancels downstream since u enters as
// u[:,d]*u[:,d]). 10 cyclic sweeps: quadratic convergence puts a 64x64
// Gaussian matrix at fp32 machine precision by ~sweep 8.
// ---------------------------------------------------------------------------
__global__ __launch_bounds__(256) void svd_u_jacobi(const float* __restrict__ Cin,
                                                    float* __restrict__ Uout) {
  __shared__ float Acol[ND][ND + 1];   // Acol[col][row], padded vs bank conflicts
  __shared__ float snorm[ND];
  __shared__ int   rnk[ND];

  const int cn  = blockIdx.x;          // 0..15
  const int tid = threadIdx.x;         // 256 threads = 8 wave32
  const float* Cm = Cin + (size_t)cn * ND * ND;

  for (int idx = tid; idx < ND * ND; idx += 256) {
    int i = idx >> 6, j = idx & 63;
    Acol[j][i] = Cm[i * ND + j];       // column-major in LDS
  }
  __syncthreads();

  const int grp = tid >> 3;            // pair group 0..31 (8 lanes each, wave-aligned)
  const int l8  = tid & 7;

  for (int sweep = 0; sweep < 10; ++sweep) {
    for (int step = 0; step < 63; ++step) {
      // round-robin tournament pairing: 32 disjoint pairs covering all 64 cols
      int p, q;
      if (grp == 0) { p = 63; q = step; }
      else {
        p = (step + grp) % 63;
        q = (step + 63 - grp) % 63;
      }
      // Gram elements of the column pair (8-lane strided + butterfly reduce)
      float app = 0.f, aqq = 0.f, apq = 0.f;
      for (int i = l8; i < ND; i += 8) {
        float ap = Acol[p][i], aq = Acol[q][i];
        app = fmaf(ap, ap, app);
        aqq = fmaf(aq, aq, aqq);
        apq = fmaf(ap, aq, apq);
      }
      for (int off = 1; off < 8; off <<= 1) {
        app += __shfl_xor(app, off, 8);
        aqq += __shfl_xor(aqq, off, 8);
        apq += __shfl_xor(apq, off, 8);
      }
      // Jacobi rotation annihilating <a_p, a_q>
      float c = 1.f, s = 0.f;
      if (fabsf(apq) > 1e-30f + 1e-14f * sqrtf(app * aqq)) {
        float tau = (aqq - app) / (2.f * apq);
        float t   = (tau >= 0.f ? 1.f : -1.f) / (fabsf(tau) + sqrtf(1.f + tau * tau));
        c = 1.f / sqrtf(1.f + t * t);
        s = t * c;
      }
      for (int i = l8; i < ND; i += 8) {
        float ap = Acol[p][i], aq = Acol[q][i];
        Acol[p][i] = c * ap - s * aq;
        Acol[q][i] = s * ap + c * aq;
      }
      __syncthreads();
    }
  }

  // singular values = column norms
  if (tid < ND) {
    float acc = 0.f;
    for (int i = 0; i < ND; ++i) { float a = Acol[tid][i]; acc = fmaf(a, a, acc); }
    snorm[tid] = sqrtf(acc);
  }
  __syncthreads();
  // stable descending rank (ties broken by original index)
  if (tid < ND) {
    float sd = snorm[tid];
    int r = 0;
    for (int e = 0; e < ND; ++e) {
      float se = snorm[e];
      r += (se > sd) || (se == sd && e < tid);
    }
    rnk[tid] = r;
  }
  __syncthreads();
  // Uout[cn][i][rank(d)] = Acol[d][i] / s_d
  float* Uo = Uout + (size_t)cn * ND * ND;
  for (int idx = tid; idx < ND * ND; idx += 256) {
    int d = idx >> 6, i = idx & 63;
    float sv  = snorm[d];
    float inv = (sv > 0.f) ? (1.f / sv) : 0.f;
    Uo[i * ND + rnk[d]] = Acol[d][i] * inv;
  }
}

// ---------------------------------------------------------------------------
// Kernel 2a: S[(b*16+o), (cn*64+d)] = D[b,cn,d] * |w[d,cn,o]|
// ---------------------------------------------------------------------------
__global__ __launch_bounds__(256) void build_S(const float* __restrict__ D,
                                               const float* __restrict__ w,
                                               float* __restrict__ S) {
  int idx = blockIdx.x * 256 + threadIdx.x;
  if (idx >= M_GEMM * K_GEMM) return;
  int col = idx & (K_GEMM - 1);        // cn*64 + d
  int row = idx >> 10;                 // b*16 + o
  int b  = row >> 4, o = row & 15;
  int cn = col >> 6, d = col & 63;
  float dv = D[((size_t)b * CN + cn) * ND + d];
  float wv = fabsf(w[((size_t)d * CN + cn) * 16 + o]);
  S[idx] = dv * wv;
}

// ---------------------------------------------------------------------------
// Kernel 2b: T[(cn*64+d), (i*64+j)] = U[cn][i][d] * U[cn][j][d]
// One block per row (1024 rows), u-column staged in LDS, float4 stores.
// ---------------------------------------------------------------------------
__global__ __launch_bounds__(256) void build_T(const float* __restrict__ U,
                                               float* __restrict__ T) {
  int row = blockIdx.x;                // cn*64 + d
  int cn = row >> 6, d = row & 63;
  __shared__ float ucol[ND];
  if (threadIdx.x < ND)
    ucol[threadIdx.x] = U[((size_t)cn * ND + threadIdx.x) * ND + d];
  __syncthreads();
  float4* Trow = (float4*)(T + (size_t)row * N_GEMM);
  for (int idx = threadIdx.x; idx < N_GEMM / 4; idx += 256) {
    int i  = idx >> 4;                 // 0..63
    int j0 = (idx & 15) * 4;           // 0,4,...,60
    float ui = ucol[i];
    float4 v;
    v.x = ui * ucol[j0 + 0];
    v.y = ui * ucol[j0 + 1];
    v.z = ui * ucol[j0 + 2];
    v.w = ui * ucol[j0 + 3];
    Trow[idx] = v;
  }
}

// ---------------------------------------------------------------------------
// Kernel 3: fp32 WMMA GEMM  out(512x4096) = S(512x1024) @ T(1024x4096)
// using V_WMMA_F32_16X16X4_F32. Register tiling: each wave owns a 32x64
// output strip (2 M-tiles x 4 N-tiles = 8 WMMAs per K-step, 64 acc VGPRs),
// so fragment traffic drops to ~192 B per WMMA. 8 waves per block arranged
// 2(M) x 4(N) cover a 64x256 block tile; grid 16x8. The L2-resident B stream
// is prefetched 32 K-rows ahead (global_prefetch_b8).
//
// Fragment layouts per CDNA5 ISA §7.12.2 (32-bit, wave32):
//   A 16x4 : lanes 0-15 -> M=lane,   v[0]=K0, v[1]=K1 ; lanes 16-31 -> K2,K3
//   B 4x16 : lanes 0-15 -> N=lane,   v[0]=K0, v[1]=K1 ; lanes 16-31 -> K2,K3
//   C/D    : VGPR r -> row r (lanes 0-15) / row r+8 (lanes 16-31), col=lane&15
// ---------------------------------------------------------------------------
__global__ __launch_bounds__(256) void wmma_gemm_f32(const float* __restrict__ A,
                                                     const float* __restrict__ B,
                                                     float* __restrict__ Cout) {
  const int wave  = threadIdx.x >> 5;
  const int lane  = threadIdx.x & 31;
  const int half  = lane >> 4;         // 0: lanes 0-15, 1: lanes 16-31
  const int l     = lane & 15;
  const int waveM = wave >> 2;         // 0..1
  const int waveN = wave & 3;          // 0..3

  const int m0 = blockIdx.y * 64 + waveM * 32;
  const int n0 = blockIdx.x * 256 + waveN * 64;

  const float* ArowLo = A + (size_t)(m0 + l) * K_GEMM;        // M-tile 0 row
  const float* ArowHi = A + (size_t)(m0 + 16 + l) * K_GEMM;   // M-tile 1 row
  const float* Bbase  = B + n0 + l;                           // B column base

  v8f acc[2][4] = {};
#pragma unroll 2
  for (int k = 0; k < K_GEMM; k += 4) {
    const int ka = k + half * 2;
    v2f a0, a1;
    a0.x = ArowLo[ka];  a0.y = ArowLo[ka + 1];
    a1.x = ArowHi[ka];  a1.y = ArowHi[ka + 1];
    v2f bf[4];
#pragma unroll
    for (int nt = 0; nt < 4; ++nt) {
      bf[nt].x = Bbase[(size_t)ka * N_GEMM + nt * 16];
      bf[nt].y = Bbase[(size_t)(ka + 1) * N_GEMM + nt * 16];
    }
    if (k + 32 < K_GEMM)  // uniform branch; prefetch B panel 32 K-rows ahead
      __builtin_prefetch(Bbase + (size_t)(k + 32 + half * 2) * N_GEMM, 0, 3);
#pragma unroll
    for (int nt = 0; nt < 4; ++nt) {
      acc[0][nt] = __builtin_amdgcn_wmma_f32_16x16x4_f32(
          false, a0, false, bf[nt], (short)0, acc[0][nt], false, false);
      acc[1][nt] = __builtin_amdgcn_wmma_f32_16x16x4_f32(
          false, a1, false, bf[nt], (short)0, acc[1][nt], false, false);
    }
  }

#pragma unroll
  for (int mt = 0; mt < 2; ++mt) {
    const int orow = m0 + mt * 16 + half * 8;
#pragma unroll
    for (int nt = 0; nt < 4; ++nt) {
      float* Co = Cout + (size_t)orow * N_GEMM + n0 + nt * 16 + l;
#pragma unroll
      for (int r = 0; r < 8; ++r) Co[(size_t)r * N_GEMM] = acc[mt][nt][r];
    }
  }
}

// ---------------------------------------------------------------------------
extern "C" void kernel_launch(void* const* d_in, const int* in_sizes, int n_in,
                              void* d_out, int out_size, void* d_ws, size_t ws_size,
                              hipStream_t stream) {
  const float* D = (const float*)d_in[0];  // (32, 2, 8, 64)
  const float* C = (const float*)d_in[1];  // (2, 8, 64, 64)
  const float* w = (const float*)d_in[2];  // (64, 2, 8, 16)
  float* out = (float*)d_out;              // (32, 16, 64, 64) == 512 x 4096

  // workspace: U (256 KB) | S (2 MB) | T (16 MB)
  char*  ws = (char*)d_ws;
  float* U  = (float*)ws;
  float* S  = (float*)(ws + (size_t)(256 << 10));
  float* T  = (float*)(ws + (size_t)(256 << 10) + (size_t)(2 << 20));

  svd_u_jacobi<<<CN, 256, 0, stream>>>(C, U);
  build_S<<<(M_GEMM * K_GEMM + 255) / 256, 256, 0, stream>>>(D, w, S);
  build_T<<<K_GEMM, 256, 0, stream>>>(U, T);
  wmma_gemm_f32<<<dim3(N_GEMM / 256, M_GEMM / 64), 256, 0, stream>>>(S, T, out);
}